// DeltaModulator_old_42923903156350
// MI455X (gfx1250) — compile-verified
//
#include <hip/hip_runtime.h>

// ---------------- problem constants ----------------
#define DELTA_V    0.02f
#define X_COLS     320
#define OUT_COLS   620
#define SCAN_LEN   300
#define TAIL_LEN   20
#define TCH        60            // timesteps per chunk
#define NCH        5             // 5 * 60 = 300
#define R          128           // rows per block
#define NT         128           // threads per block (thread t scans row t)
#define T4         (TCH / 4)     // float4 groups per row per chunk (15)
#define LSTRIDE    68            // LDS dwords/row: 16B-aligned, 2-way bank conflict only

// ---------------- CDNA5 async global->LDS path ----------------
#if defined(__AMDGCN__) && __has_builtin(__builtin_amdgcn_global_load_async_to_lds_b128)
#define HAVE_ASYNC_LDS 1
#else
#define HAVE_ASYNC_LDS 0
#if defined(__AMDGCN__)
#warning "gfx1250 global_load_async_to_lds_b128 builtin NOT available; using sync float4 fallback"
#endif
#endif

typedef int   v4i __attribute__((vector_size(16)));
typedef float v4f __attribute__((ext_vector_type(4)));

__device__ __forceinline__ void wait_async_zero() {
#if HAVE_ASYNC_LDS
#if __has_builtin(__builtin_amdgcn_s_wait_asynccnt)
    __builtin_amdgcn_s_wait_asynccnt(0);
#else
    asm volatile("s_wait_asynccnt 0x0" ::: "memory");
#endif
#endif
}

// Copy 16 bytes from global to LDS (async DMA on CDNA5, sync fallback otherwise).
__device__ __forceinline__ void copy16_g2l(const float* __restrict__ g, float* l) {
#if HAVE_ASYNC_LDS
    __builtin_amdgcn_global_load_async_to_lds_b128(
        (__attribute__((address_space(1))) v4i*)(g),
        (__attribute__((address_space(3))) v4i*)(l),
        /*imm offset*/ 0, /*cpol*/ 0);
#else
    *(v4f*)l = *(const v4f*)g;
#endif
}

// Stage one (R x TCH) input chunk into an LDS buffer; 128-row tile is
// contiguous in global memory, each row-chunk a contiguous 240B segment.
__device__ __forceinline__ void stage_chunk(const float* __restrict__ xb,
                                            float* __restrict__ sbuf,
                                            int k, int tid) {
    const float* src = xb + k * TCH;
    for (int idx = tid; idx < R * T4; idx += NT) {
        int row = idx / T4;
        int c4  = idx - row * T4;
        copy16_g2l(src + (long)row * X_COLS + (c4 << 2),
                   sbuf + row * LSTRIDE + (c4 << 2));
    }
}

__global__ __launch_bounds__(NT) void delta_mod_kernel(const float* __restrict__ x,
                                                       float* __restrict__ out) {
    __shared__ float s_in[2][R * LSTRIDE];   // double-buffered input chunks
    __shared__ float s_up[R * LSTRIDE];      // UP flags for one chunk
    __shared__ float s_dn[R * LSTRIDE];      // DN flags for one chunk

    const int  tid     = threadIdx.x;
    const long rowbase = (long)blockIdx.x * R;
    const float* xb = x   + rowbase * X_COLS;
    float*       ob = out + rowbase * OUT_COLS;

    // ---- tail: out[:,600:620] = x[:,300:320], vectorized + non-temporal ----
    // 5 float4 per row, 128 rows -> 640 items, 5 per thread.
    for (int idx = tid; idx < R * (TAIL_LEN / 4); idx += NT) {
        int row = idx / (TAIL_LEN / 4);
        int c4  = idx - row * (TAIL_LEN / 4);
        v4f v = __builtin_nontemporal_load(
            (const v4f*)(xb + (long)row * X_COLS + SCAN_LEN + (c4 << 2)));
        __builtin_nontemporal_store(
            v, (v4f*)(ob + (long)row * OUT_COLS + 2 * SCAN_LEN + (c4 << 2)));
    }

    // ---- prologue: stage chunk 0 ----
    stage_chunk(xb, s_in[0], 0, tid);
    wait_async_zero();
    __syncthreads();

    float dc = 0.0f;   // per-row scan state, carried across chunks

    for (int k = 0; k < NCH; ++k) {
        const int cur = k & 1;
        const int nxt = cur ^ 1;

        // overlap: kick off async staging of the next chunk
        if (k + 1 < NCH)
            stage_chunk(xb, s_in[nxt], k + 1, tid);

        // ---- sequential delta-modulator scan: thread t owns row t ----
        {
            const float* rin = &s_in[cur][tid * LSTRIDE];
            float*       rup = &s_up[tid * LSTRIDE];
            float*       rdn = &s_dn[tid * LSTRIDE];
            for (int i4 = 0; i4 < T4; ++i4) {
                v4f xi4 = *(const v4f*)(rin + (i4 << 2));   // ds_load_b128
                v4f u4, d4;
                #pragma unroll
                for (int j = 0; j < 4; ++j) {
                    float xi = xi4[j];
                    bool up = xi > dc + DELTA_V;
                    bool dn = xi < dc - DELTA_V;
                    u4[j] = up ? 1.0f : 0.0f;
                    d4[j] = dn ? 1.0f : 0.0f;
                    dc = (up || dn) ? xi : dc;
                }
                *(v4f*)(rup + (i4 << 2)) = u4;              // ds_store_b128
                *(v4f*)(rdn + (i4 << 2)) = d4;              // ds_store_b128
            }
        }
        __syncthreads();   // UP/DN tiles complete

        // ---- vectorized, non-temporal store of UP/DN chunk ----
        // 15 float4 per row -> 240B contiguous segments per row.
        for (int idx = tid; idx < R * T4; idx += NT) {
            int row = idx / T4;
            int c4  = idx - row * T4;
            v4f u = *(const v4f*)(&s_up[row * LSTRIDE + (c4 << 2)]);
            v4f d = *(const v4f*)(&s_dn[row * LSTRIDE + (c4 << 2)]);
            float* o = ob + (long)row * OUT_COLS + k * TCH + (c4 << 2);
            __builtin_nontemporal_store(u, (v4f*)(o));
            __builtin_nontemporal_store(d, (v4f*)(o + SCAN_LEN));
        }

        // next input buffer ready; also fences s_up/s_dn + s_in[cur] reuse
        wait_async_zero();
        __syncthreads();
    }
}

extern "C" void kernel_launch(void* const* d_in, const int* in_sizes, int n_in,
                              void* d_out, int out_size, void* d_ws, size_t ws_size,
                              hipStream_t stream) {
    const float* x   = (const float*)d_in[0];
    float*       out = (float*)d_out;

    const long rows_total = (long)in_sizes[0] / X_COLS;   // 2048*128 = 262144
    const int  nblocks    = (int)(rows_total / R);        // 2048

    delta_mod_kernel<<<dim3(nblocks), dim3(NT), 0, stream>>>(x, out);
}